// DistanceConstraint_14474039787802
// MI455X (gfx1250) — compile-verified
//
#include <hip/hip_runtime.h>

typedef __attribute__((ext_vector_type(2))) float v2f;
typedef __attribute__((ext_vector_type(8))) float v8f;

#define B_DIM 8
#define N_DIM 2048
#define D_DIM 512
#define WAVES_PER_BLOCK 8
// Triangular enumeration of 16-row x 64-col blocks:
//   col-blocks cb = 0..31 (64 cols each), row tiles ti = 0..(4*cb+3)
//   prefix(cb) = 2*cb*(cb+1); total per batch = 2*32*33 = 2112
#define TRI_BLOCKS_PER_BATCH 2112
#define TOTAL_WAVES (B_DIM * TRI_BLOCKS_PER_BATCH)       // 16896
#define GRID_BLOCKS (TOTAL_WAVES / WAVES_PER_BLOCK)      // 2112

// ---------------------------------------------------------------------------
// Kernel 1: per-row stats. One wave per row (32 lanes x 16 floats = D=512).
// invn[row] = 1 / max(||x||, 1e-12) ; sqn[row] = ||x * invn||^2
// ---------------------------------------------------------------------------
__global__ void row_stats_kernel(const float* __restrict__ emb,
                                 float* __restrict__ invn,
                                 float* __restrict__ sqn) {
  int wave = threadIdx.x >> 5;
  int lane = threadIdx.x & 31;
  int row  = blockIdx.x * WAVES_PER_BLOCK + wave;   // 0 .. B*N-1
  const float* p = emb + (size_t)row * D_DIM + lane * 16;
  float s = 0.f;
#pragma unroll
  for (int i = 0; i < 4; ++i) {
    float4 v = ((const float4*)p)[i];
    s += v.x * v.x + v.y * v.y + v.z * v.z + v.w * v.w;
  }
#pragma unroll
  for (int off = 16; off > 0; off >>= 1)
    s += __shfl_xor(s, off, 32);
  if (lane == 0) {
    float nrm = sqrtf(s);
    float iv  = 1.f / fmaxf(nrm, 1e-12f);
    invn[row] = iv;
    sqn[row]  = s * iv * iv;
  }
}

// ---------------------------------------------------------------------------
// Kernel 2: one wave per 16x64 strip of the (upper-triangular) tile grid.
// 4 accumulators share one A fragment per k-step: 5 b64 loads / 4 WMMAs.
// Off-diagonal 16x16 sub-tiles weighted x2 (symmetry), diagonal x1,
// below-diagonal x0.
// ---------------------------------------------------------------------------
__global__ void tile_loss_kernel(const float* __restrict__ emb,
                                 const float* __restrict__ coords,
                                 const float* __restrict__ mask,
                                 const float* __restrict__ invn,
                                 const float* __restrict__ sqn,
                                 float* __restrict__ partials) {
  __shared__ float wsum[WAVES_PER_BLOCK];

  int wave = threadIdx.x >> 5;
  int lane = threadIdx.x & 31;
  int w    = blockIdx.x * WAVES_PER_BLOCK + wave;
  int b    = w / TRI_BLOCKS_PER_BATCH;
  int t    = w - b * TRI_BLOCKS_PER_BATCH;

  // decode triangular block index: cb s.t. 2*cb*(cb+1) <= t < 2*(cb+1)*(cb+2)
  int cb = (int)((sqrtf((float)(1 + 2 * t)) - 1.0f) * 0.5f);
  while (2 * (cb + 1) * (cb + 2) <= t) ++cb;
  while (2 * cb * (cb + 1) > t) --cb;
  int ti = t - 2 * cb * (cb + 1);                   // 0 .. 4*cb+3

  int rowBase = ti * 16;
  int colBase = cb * 64;
  int lmod  = lane & 15;
  int lhalf = lane >> 4;

  // A fragment: lane holds row (rowBase+lmod), K = k + lhalf*2 + {0,1}
  const float* Arow = emb + ((size_t)b * N_DIM + rowBase + lmod) * D_DIM + lhalf * 2;
  const float* Brow0 = emb + ((size_t)b * N_DIM + colBase +  0 + lmod) * D_DIM + lhalf * 2;
  const float* Brow1 = emb + ((size_t)b * N_DIM + colBase + 16 + lmod) * D_DIM + lhalf * 2;
  const float* Brow2 = emb + ((size_t)b * N_DIM + colBase + 32 + lmod) * D_DIM + lhalf * 2;
  const float* Brow3 = emb + ((size_t)b * N_DIM + colBase + 48 + lmod) * D_DIM + lhalf * 2;

  v8f c0 = {}, c1 = {}, c2 = {}, c3 = {};
#pragma unroll 4
  for (int k = 0; k < D_DIM; k += 4) {
    v2f a  = *(const v2f*)(Arow + k);
    v2f b0 = *(const v2f*)(Brow0 + k);
    v2f b1 = *(const v2f*)(Brow1 + k);
    v2f b2 = *(const v2f*)(Brow2 + k);
    v2f b3 = *(const v2f*)(Brow3 + k);
    c0 = __builtin_amdgcn_wmma_f32_16x16x4_f32(false, a, false, b0, (short)0, c0, false, false);
    c1 = __builtin_amdgcn_wmma_f32_16x16x4_f32(false, a, false, b1, (short)0, c1, false, false);
    c2 = __builtin_amdgcn_wmma_f32_16x16x4_f32(false, a, false, b2, (short)0, c2, false, false);
    c3 = __builtin_amdgcn_wmma_f32_16x16x4_f32(false, a, false, b3, (short)0, c3, false, false);
  }

  // ---- epilogue: C/D layout -> lane covers N = lmod, M = r + lhalf*8 ----
  // Row-side stats loaded once, shared across the 4 sub-tiles.
  int mbase = rowBase + (lhalf ? 8 : 0);
  float invn_i[8], sq_i[8], m_i[8], cix[8], ciy[8], ciz[8];
#pragma unroll
  for (int r = 0; r < 8; ++r) {
    size_t bi = (size_t)b * N_DIM + mbase + r;
    invn_i[r] = invn[bi];
    sq_i[r]   = sqn[bi];
    m_i[r]    = mask[bi];
    const float* ci = coords + bi * 3;
    cix[r] = ci[0]; ciy[r] = ci[1]; ciz[r] = ci[2];
  }

  float acc = 0.f;
  auto subtile = [&](const v8f& c, int s) {
    int tj = cb * 4 + s;
    float wgt = (tj < ti) ? 0.f : ((tj == ti) ? 1.f : 2.f);
    int    j  = colBase + s * 16 + lmod;
    size_t bj = (size_t)b * N_DIM + j;
    float invn_j = invn[bj];
    float sq_j   = sqn[bj];
    float m_jw   = mask[bj] * wgt;
    const float* cj = coords + bj * 3;
    float cjx = cj[0], cjy = cj[1], cjz = cj[2];
    float a2 = 0.f;
#pragma unroll
    for (int r = 0; r < 8; ++r) {
      float dx = cix[r] - cjx, dy = ciy[r] - cjy, dz = ciz[r] - cjz;
      float cd2 = dx * dx + dy * dy + dz * dz;        // coord_dist^2
      float sd  = c[r] * invn_i[r] * invn_j;          // normalized dot
      float d2  = fmaxf(sq_i[r] + sq_j - 2.f * sd, 0.f);
      float l   = fmaxf(sqrtf(d2) - 1.0f, 0.f);       // relu(d - MARGIN)
      a2 += (cd2 < 100.0f) ? (l * m_i[r] * m_jw) : 0.f;
    }
    acc += a2;
  };
  subtile(c0, 0); subtile(c1, 1); subtile(c2, 2); subtile(c3, 3);

  // fixed-order wave reduction
#pragma unroll
  for (int off = 16; off > 0; off >>= 1)
    acc += __shfl_xor(acc, off, 32);
  if (lane == 0) wsum[wave] = acc;
  __syncthreads();
  if (threadIdx.x == 0) {
    float ssum = 0.f;
#pragma unroll
    for (int v = 0; v < WAVES_PER_BLOCK; ++v) ssum += wsum[v];
    partials[blockIdx.x] = ssum;
  }
}

// ---------------------------------------------------------------------------
// Kernel 3: deterministic final reduction + valid_count from mask.
// valid_count = sum_b (sum_n mask[b,n])^2 ; out = total / (valid + 1e-8)
// ---------------------------------------------------------------------------
__global__ void finalize_kernel(const float* __restrict__ partials,
                                const float* __restrict__ mask,
                                float* __restrict__ out) {
  __shared__ float red[256];
  int tid = threadIdx.x;

  float s = 0.f;
  for (int i = tid; i < GRID_BLOCKS; i += 256) s += partials[i];
  red[tid] = s;
  __syncthreads();
  for (int off = 128; off > 0; off >>= 1) {
    if (tid < off) red[tid] += red[tid + off];
    __syncthreads();
  }
  float total = red[0];
  __syncthreads();

  float vsum = 0.f;
  for (int b = 0; b < B_DIM; ++b) {
    float ms = 0.f;
    for (int i = tid; i < N_DIM; i += 256) ms += mask[(size_t)b * N_DIM + i];
    red[tid] = ms;
    __syncthreads();
    for (int off = 128; off > 0; off >>= 1) {
      if (tid < off) red[tid] += red[tid + off];
      __syncthreads();
    }
    if (tid == 0) { float sb = red[0]; vsum += sb * sb; }
    __syncthreads();
  }
  if (tid == 0) out[0] = total / (vsum + 1e-8f);
}

// ---------------------------------------------------------------------------
extern "C" void kernel_launch(void* const* d_in, const int* in_sizes, int n_in,
                              void* d_out, int out_size, void* d_ws, size_t ws_size,
                              hipStream_t stream) {
  const float* emb    = (const float*)d_in[0];   // [8,2048,512] f32
  const float* coords = (const float*)d_in[1];   // [8,2048,3]   f32
  const float* mask   = (const float*)d_in[2];   // [8,2048]     f32

  float* wsf      = (float*)d_ws;
  float* invn     = wsf;                          // B*N floats
  float* sqn      = wsf + B_DIM * N_DIM;          // B*N floats
  float* partials = wsf + 2 * B_DIM * N_DIM;      // GRID_BLOCKS floats

  row_stats_kernel<<<B_DIM * N_DIM / WAVES_PER_BLOCK, 256, 0, stream>>>(emb, invn, sqn);
  tile_loss_kernel<<<GRID_BLOCKS, 256, 0, stream>>>(emb, coords, mask, invn, sqn, partials);
  finalize_kernel<<<1, 256, 0, stream>>>(partials, mask, (float*)d_out);
}